// AlphaBetaFilter_14396730377048
// MI455X (gfx1250) — compile-verified
//
#include <hip/hip_runtime.h>
#include <math.h>

// Problem constants (match reference)
#define B_      32
#define T_      4096
#define C_      128
#define NCHUNK  32
#define CLEN    (T_ / NCHUNK)   // 128 timesteps per chunk
#define SUB     16              // subtile rows (double-buffered)
#define NSUB    (CLEN / SUB)    // 8
#define LANES   32              // one wave per block, lane == channel-in-quarter

// ---------- CDNA5 async global->LDS helpers (gfx1250) ----------

// 32-bit LDS byte address of a __shared__ object (addrspace(3) offset).
__device__ __forceinline__ unsigned lds_lo32(const void* p) {
  return (unsigned)(unsigned long long)(const __attribute__((address_space(3))) char*)p;
}

// ASYNCcnt-tracked DMA: 16B per lane, global -> LDS, no VGPR data.
__device__ __forceinline__ void async_b128(unsigned lds_byte, const float* g) {
  asm volatile("global_load_async_to_lds_b128 %0, %1, off"
               :: "v"(lds_byte), "v"(g)
               : "memory");
}

__device__ __forceinline__ void wait_async_le4() {
  asm volatile("s_wait_asynccnt 0x4" ::: "memory");
}
__device__ __forceinline__ void wait_async_0() {
  asm volatile("s_wait_asynccnt 0x0" ::: "memory");
}

__device__ __forceinline__ float sig_clamp(float z) {
  float s = 1.0f / (1.0f + expf(-z));
  return fminf(fmaxf(s, 1.0e-4f), 1.0f - 1.0e-4f);
}

// One subtile = 16 rows x 32 channels (this wave's quarter) = 2 KiB.
// 4 async b128 instructions per wave: lane covers rows (lane>>3)+{0,4,8,12},
// columns c0 + (lane&7)*4 .. +3.  128B contiguous per 8-lane group.
__device__ __forceinline__ void issue_subtile(const float* xsub, int c0,
                                              unsigned lds_base, int lane) {
  const int tt = lane >> 3;          // 0..3
  const int cc = (lane & 7) << 2;    // 0,4,...,28
  const float* g  = xsub + (size_t)tt * C_ + (c0 + cc);
  const unsigned l = lds_base + (unsigned)((tt * LANES + cc) * 4);
  async_b128(l,                  g);
  async_b128(l + 4*LANES*4,      g + (size_t)4  * C_);
  async_b128(l + 8*LANES*4,      g + (size_t)8  * C_);
  async_b128(l + 12*LANES*4,     g + (size_t)12 * C_);
}

// ---------- Pass kernel (WRITE=0: per-chunk zero-state scan -> ws;
//                         WRITE=1: replay from true start state -> out) ----
template <int WRITE>
__global__ __launch_bounds__(LANES)
void ab_filter_pass(const float* __restrict__ x,
                    const float* __restrict__ logit_a,
                    const float* __restrict__ logit_b,
                    float2* __restrict__ ws,
                    float* __restrict__ out) {
  const int lane = threadIdx.x;      // channel within quarter
  const int q    = blockIdx.x;       // channel quarter 0..3
  const int j    = blockIdx.y;       // chunk 0..31
  const int b    = blockIdx.z;       // batch
  const int c0   = q * LANES;
  const int c    = c0 + lane;

  const float av = sig_clamp(logit_a[c]);
  const float bv = sig_clamp(logit_b[c]);
  const float ab = av * bv;

  __shared__ __align__(16) float tile[2 * SUB * LANES];  // 4 KiB double buffer
  const unsigned tb       = lds_lo32(tile);
  const unsigned bufbytes = SUB * LANES * 4;

  const float* xchunk = x   + ((size_t)b * T_ + (size_t)j * CLEN) * C_;
  float*       ochunk = out + ((size_t)b * T_ + (size_t)j * CLEN) * C_;

  // Prologue: subtile 0 -> buffer 0 (ASYNCcnt += 4)
  issue_subtile(xchunk, c0, tb, lane);

  float L = 0.0f, S = 0.0f;
  if (WRITE && j > 0) {
    // True state at t = j*CLEN - 1, produced by the combine kernel.
    const float2 st = ws[((size_t)b * NCHUNK + j) * C_ + c];
    L = st.x; S = st.y;
  }

  for (int s = 0; s < NSUB; ++s) {
    if (s + 1 < NSUB) {
      issue_subtile(xchunk + (size_t)(s + 1) * SUB * C_, c0,
                    tb + (unsigned)((s + 1) & 1) * bufbytes, lane);
      wait_async_le4();   // subtile s landed; s+1 still in flight
    } else {
      wait_async_0();
    }
    const float* buf = tile + (s & 1) * (SUB * LANES);

    if (j == 0 && s == 0) {
      // t = 0: L0 = x0, slope = 0; out[:,0,:] = L0
      L = buf[lane];
      S = 0.0f;
      if (WRITE) ochunk[lane] = L;
#pragma unroll
      for (int tt = 1; tt < SUB; ++tt) {
        const float xv   = buf[tt * LANES + lane];   // ds_load_b32
        const float pred = L + S;
        const float e    = xv - pred;
        L = fmaf(av, e, pred);                        // L = pred + a*(x-pred)
        S = fmaf(ab, e, S);                           // s += a*b*(x-pred)
        if (WRITE) ochunk[(size_t)tt * C_ + lane] = L;
      }
    } else {
      const int t0 = s * SUB;
#pragma unroll
      for (int tt = 0; tt < SUB; ++tt) {
        const float xv   = buf[tt * LANES + lane];
        const float pred = L + S;
        const float e    = xv - pred;
        L = fmaf(av, e, pred);
        S = fmaf(ab, e, S);
        if (WRITE) ochunk[(size_t)(t0 + tt) * C_ + lane] = L;
      }
    }
  }

  if (!WRITE) {
    // Chunk 0: true end state e_0.  Chunk j>0: zero-state contribution d_j.
    ws[((size_t)b * NCHUNK + j) * C_ + c] = make_float2(L, S);
  }
}

// ---------- Combine: e_j = A^CLEN * e_{j-1} + d_j ; store start states ----
__global__ __launch_bounds__(C_)
void ab_combine(const float* __restrict__ logit_a,
                const float* __restrict__ logit_b,
                float2* __restrict__ ws) {
  const int c = threadIdx.x;
  const int b = blockIdx.x;

  const float av = sig_clamp(logit_a[c]);
  const float bv = sig_clamp(logit_b[c]);
  const float ab = av * bv;

  // A = [[1-a, 1-a], [-ab, 1-ab]];  A^128 via 7 squarings.
  float m00 = 1.0f - av, m01 = 1.0f - av;
  float m10 = -ab,       m11 = 1.0f - ab;
#pragma unroll
  for (int i = 0; i < 7; ++i) {
    const float n00 = m00 * m00 + m01 * m10;
    const float n01 = m00 * m01 + m01 * m11;
    const float n10 = m10 * m00 + m11 * m10;
    const float n11 = m10 * m01 + m11 * m11;
    m00 = n00; m01 = n01; m10 = n10; m11 = n11;
  }

  float2* row = ws + (size_t)b * NCHUNK * C_ + c;
  float2 e = row[0];                       // e_0 (chunk 0 true end state)
  for (int jj = 1; jj < NCHUNK; ++jj) {
    const float2 d = row[(size_t)jj * C_]; // d_jj
    row[(size_t)jj * C_] = e;              // start state for chunk jj (in place)
    const float eL = m00 * e.x + m01 * e.y + d.x;
    const float eS = m10 * e.x + m11 * e.y + d.y;
    e.x = eL; e.y = eS;
  }
}

// ---------- Launch ----------
extern "C" void kernel_launch(void* const* d_in, const int* in_sizes, int n_in,
                              void* d_out, int out_size, void* d_ws, size_t ws_size,
                              hipStream_t stream) {
  (void)in_sizes; (void)n_in; (void)out_size; (void)ws_size;
  const float* x  = (const float*)d_in[0];
  const float* la = (const float*)d_in[1];
  const float* lb = (const float*)d_in[2];
  float*  out = (float*)d_out;
  float2* ws  = (float2*)d_ws;   // needs B*NCHUNK*C*8 = 1 MiB

  dim3 grid(C_ / LANES, NCHUNK, B_);   // 4 x 32 x 32 = 4096 one-wave blocks
  ab_filter_pass<0><<<grid, LANES, 0, stream>>>(x, la, lb, ws, out);
  ab_combine<<<dim3(B_), C_, 0, stream>>>(la, lb, ws);
  ab_filter_pass<1><<<grid, LANES, 0, stream>>>(x, la, lb, ws, out);
}